// PFNLayer_33981781246289
// MI455X (gfx1250) — compile-verified
//
#include <hip/hip_runtime.h>

typedef __attribute__((ext_vector_type(2))) float v2f;
typedef __attribute__((ext_vector_type(8))) float v8f;

#define U_PILLARS 40000
#define NPTS      32
#define CIN       9
#define COUT      64
#define ROWS      (U_PILLARS * NPTS)   // 1,280,000 = 256 * 5000
#define NTRI      45                   // 9*10/2 upper-triangular moments

// workspace layout (floats)
#define WS_SUM    0    // 9   : per-channel input sums
#define WS_TRI    9    // 45  : upper-tri second moments of inputs
#define WS_SCALE  64   // 64  : per-output fused BN scale
#define WS_BIAS   128  // 64  : per-output fused BN bias

// ---------------------------------------------------------------------------
// Zero the moment accumulators (ws is poisoned by the harness; graph-safe).
// ---------------------------------------------------------------------------
__global__ void pfn_init_ws(float* __restrict__ ws) {
    int t = threadIdx.x;
    if (t < 64) ws[t] = 0.0f;   // covers WS_SUM + WS_TRI (54 floats)
}

// ---------------------------------------------------------------------------
// Pass 1: accumulate sum[c] and sum of in[c1]*in[c2] (c1<=c2) over all rows.
// 1000 blocks x 256 threads, each block does exactly 5 tiles of 256 rows.
// ---------------------------------------------------------------------------
__global__ __launch_bounds__(256) void pfn_moments(const float* __restrict__ in,
                                                   float* __restrict__ ws) {
    __shared__ float sdata[256 * CIN];
    float s[CIN];
    float t[NTRI];
#pragma unroll
    for (int i = 0; i < CIN; ++i) s[i] = 0.0f;
#pragma unroll
    for (int i = 0; i < NTRI; ++i) t[i] = 0.0f;

    const int tid = threadIdx.x;
    for (long long base = (long long)blockIdx.x * 256; base < ROWS;
         base += 256LL * gridDim.x) {
        const float* src = in + base * CIN;
        // coalesced stage: 2304 contiguous floats
#pragma unroll
        for (int i = 0; i < CIN; ++i) sdata[tid + 256 * i] = src[tid + 256 * i];
        __syncthreads();

        float v[CIN];
#pragma unroll
        for (int c = 0; c < CIN; ++c) v[c] = sdata[tid * CIN + c]; // stride 9: bank-conflict-free
#pragma unroll
        for (int c = 0; c < CIN; ++c) s[c] += v[c];
        int k = 0;
#pragma unroll
        for (int c1 = 0; c1 < CIN; ++c1)
#pragma unroll
            for (int c2 = c1; c2 < CIN; ++c2) t[k++] += v[c1] * v[c2];
        __syncthreads();
    }

    const int lane = tid & 31;
#pragma unroll
    for (int c = 0; c < CIN; ++c) {
        float r = s[c];
        for (int off = 16; off; off >>= 1) r += __shfl_xor(r, off, 32);
        if (lane == 0) atomicAdd(&ws[WS_SUM + c], r);
    }
#pragma unroll
    for (int i = 0; i < NTRI; ++i) {
        float r = t[i];
        for (int off = 16; off; off >>= 1) r += __shfl_xor(r, off, 32);
        if (lane == 0) atomicAdd(&ws[WS_TRI + i], r);
    }
}

// ---------------------------------------------------------------------------
// Derive fused BN scale/bias per output channel from the analytic moments:
//   mean_o = W_o . m,  E[x_o^2] = W_o^T S W_o,  var = E[x^2] - mean^2
//   scale = gamma * rsqrt(var+eps), bias = beta - mean*scale
// ---------------------------------------------------------------------------
__global__ void pfn_stats(const float* __restrict__ W,
                          const float* __restrict__ gamma,
                          const float* __restrict__ beta,
                          float* __restrict__ ws) {
    int o = threadIdx.x;
    if (o >= COUT) return;
    const float invN = 1.0f / (float)ROWS;
    float w[CIN];
#pragma unroll
    for (int c = 0; c < CIN; ++c) w[c] = W[o * CIN + c];

    float mean = 0.0f;
#pragma unroll
    for (int c = 0; c < CIN; ++c) mean += w[c] * ws[WS_SUM + c];
    mean *= invN;

    float ex2 = 0.0f;
#pragma unroll
    for (int c1 = 0; c1 < CIN; ++c1) {
#pragma unroll
        for (int c2 = 0; c2 < CIN; ++c2) {
            int a = c1 < c2 ? c1 : c2;
            int b = c1 < c2 ? c2 : c1;
            int idx = a * CIN - (a * (a - 1)) / 2 + (b - a);
            ex2 += w[c1] * w[c2] * ws[WS_TRI + idx];
        }
    }
    ex2 *= invN;
    float var = ex2 - mean * mean;
    float sc = gamma[o] * rsqrtf(var + 1e-3f);
    ws[WS_SCALE + o] = sc;
    ws[WS_BIAS + o]  = beta[o] - mean * sc;
}

// ---------------------------------------------------------------------------
// Pass 2: fused GEMM (WMMA f32 16x16x4) + BN + ReLU + max over 32 points.
// Block = 256 threads = 8 waves; each wave owns one pillar (32 rows x 64 out).
// LDS rows padded to 12 floats: zero K-padding for K=9->12 and clean
// 8B-aligned ds_load_b64 fragment reads.
// ---------------------------------------------------------------------------
__global__ __launch_bounds__(256) void pfn_gemm_bn_max(const float* __restrict__ in,
                                                       const float* __restrict__ W,
                                                       const float* __restrict__ ws,
                                                       float* __restrict__ out) {
    __shared__ float lds_in[256 * 12];   // 8 pillars x 32 rows, padded
    __shared__ float lds_w [COUT * 12];  // W rows, padded

    const int tid = threadIdx.x;
    const long long base = (long long)blockIdx.x * 256;   // first row of block
    const float* src = in + base * CIN;

    // coalesced stage of 2304 contiguous floats, de-interleave into padded rows
#pragma unroll
    for (int i = 0; i < CIN; ++i) {
        int idx = tid + 256 * i;
        int row = idx / CIN;
        int c   = idx % CIN;
        lds_in[row * 12 + c] = src[idx];
    }
    lds_in[tid * 12 +  9] = 0.0f;
    lds_in[tid * 12 + 10] = 0.0f;
    lds_in[tid * 12 + 11] = 0.0f;
    if (tid < COUT) {
#pragma unroll
        for (int c = 0; c < CIN; ++c) lds_w[tid * 12 + c] = W[tid * CIN + c];
        lds_w[tid * 12 +  9] = 0.0f;
        lds_w[tid * 12 + 10] = 0.0f;
        lds_w[tid * 12 + 11] = 0.0f;
    }
    __syncthreads();

    const int wave    = tid >> 5;
    const int lane    = tid & 31;
    const int laneRow = lane & 15;  // M (A) / N (B) index within 16-wide tile
    const int laneHi  = lane >> 4;  // selects K sub-pair {0,1} vs {2,3}

    v8f acc[2][4];
#pragma unroll
    for (int mt = 0; mt < 2; ++mt)
#pragma unroll
        for (int nt = 0; nt < 4; ++nt) acc[mt][nt] = (v8f){};

#pragma unroll
    for (int kc = 0; kc < 3; ++kc) {       // K = 9 padded to 12 = 3 x 4
        v2f a[2], b[4];
#pragma unroll
        for (int mt = 0; mt < 2; ++mt)     // A 16x4: lane=M+16*(K/2), vgpr=K%2
            a[mt] = *(const v2f*)&lds_in[(wave * 32 + mt * 16 + laneRow) * 12 +
                                         kc * 4 + laneHi * 2];
#pragma unroll
        for (int nt = 0; nt < 4; ++nt)     // B 4x16: lane=N+16*(K/2), vgpr=K%2
            b[nt] = *(const v2f*)&lds_w[(nt * 16 + laneRow) * 12 +
                                        kc * 4 + laneHi * 2];
#pragma unroll
        for (int mt = 0; mt < 2; ++mt)
#pragma unroll
            for (int nt = 0; nt < 4; ++nt)
                acc[mt][nt] = __builtin_amdgcn_wmma_f32_16x16x4_f32(
                    false, a[mt], false, b[nt], (short)0, acc[mt][nt],
                    false, false);
    }

    // BN + ReLU + max over 32 points.
    // C/D layout: vgpr r -> M=r (lanes 0-15) / M=8+r (lanes 16-31), N=lane%16.
    const float* scale = ws + WS_SCALE;
    const float* bias  = ws + WS_BIAS;
    const int u = blockIdx.x * 8 + wave;   // pillar index
#pragma unroll
    for (int nt = 0; nt < 4; ++nt) {
        float sc = scale[nt * 16 + laneRow];
        float bi = bias [nt * 16 + laneRow];
        float m = 0.0f;                     // ReLU outputs are >= 0
#pragma unroll
        for (int mt = 0; mt < 2; ++mt)
#pragma unroll
            for (int r = 0; r < 8; ++r)
                m = fmaxf(m, fmaf(acc[mt][nt][r], sc, bi));
        // combine the two half-wave row groups (M 0-7/16-23 vs 8-15/24-31)
        m = fmaxf(m, __shfl_xor(m, 16, 32));
        if (lane < 16) out[u * COUT + nt * 16 + lane] = m;
    }
}

// ---------------------------------------------------------------------------
extern "C" void kernel_launch(void* const* d_in, const int* in_sizes, int n_in,
                              void* d_out, int out_size, void* d_ws, size_t ws_size,
                              hipStream_t stream) {
    (void)in_sizes; (void)n_in; (void)out_size; (void)ws_size;
    const float* in    = (const float*)d_in[0];
    const float* W     = (const float*)d_in[1];
    const float* gamma = (const float*)d_in[2];
    const float* beta  = (const float*)d_in[3];
    float* out = (float*)d_out;
    float* ws  = (float*)d_ws;

    pfn_init_ws<<<1, 64, 0, stream>>>(ws);
    pfn_moments<<<1000, 256, 0, stream>>>(in, ws);
    pfn_stats<<<1, 64, 0, stream>>>(W, gamma, beta, ws);
    pfn_gemm_bn_max<<<U_PILLARS / 8, 256, 0, stream>>>(in, W, ws, out);
}